// GELU220_23648089932112
// MI455X (gfx1250) — compile-verified
//
#include <hip/hip_runtime.h>
#include <hip/hip_bf16.h>
#include <stdint.h>

#define DCH 2048          // channel dim (from reference: D = 2048)
#define THREADS 256

#if defined(__gfx1250__) && __has_builtin(__builtin_amdgcn_tensor_load_to_lds)
#define HAVE_TDM 1
#endif

typedef unsigned int u32;
typedef u32   v4u __attribute__((ext_vector_type(4)));
typedef int   v8i __attribute__((ext_vector_type(8)));
typedef int   v4i __attribute__((ext_vector_type(4)));
typedef float v4f __attribute__((ext_vector_type(4)));

__device__ __forceinline__ float gelu_f(float v) {
    const float c = 0.7978845608028654f; // sqrt(2/pi)
    float t = tanhf(c * (v + 0.044715f * v * v * v));
    return 0.5f * v * (1.0f + t);
}

// ---------------------------------------------------------------------------
// Kernel 1: per-channel partial sums / sums-of-squares over a chunk of rows.
// grid = (DCH/1024, nchunks), block = 256 threads, v4f per thread.
// Deterministic: each chunk writes its own partial slot; no atomics.
// ---------------------------------------------------------------------------
__global__ void colstats_kernel(const float* __restrict__ x,
                                float* __restrict__ ws,
                                int rowsPerChunk) {
    const int tid = threadIdx.x;
    const int d4  = blockIdx.x * THREADS + tid;       // v4f index in [0, DCH/4)
    const int row0 = blockIdx.y * rowsPerChunk;
    v4f s = {0.f, 0.f, 0.f, 0.f};
    v4f q = {0.f, 0.f, 0.f, 0.f};
    for (int r = 0; r < rowsPerChunk; ++r) {
        const v4f v = ((const v4f*)(x + (size_t)(row0 + r) * DCH))[d4];
        s += v;
        q += v * v;
    }
    float* base = ws + (size_t)blockIdx.y * (2 * DCH);
    ((v4f*)base)[d4]         = s;
    ((v4f*)(base + DCH))[d4] = q;
}

// ---------------------------------------------------------------------------
// Kernel 2: single block. Reduce chunk partials (fixed order), build fused
// per-channel coefficients a[d], b[d], normalized ema_n[d], and scalars.
// ws layout (floats): [0, nchunks*2*DCH) partials
//                     [CH+0, CH+DCH)      a
//                     [CH+DCH, CH+2DCH)   b
//                     [CH+2DCH, CH+3DCH)  ema_n
//                     [CH+3DCH + {0,1,2}] beta_up, beta_dn, tau
// ---------------------------------------------------------------------------
__global__ void finalize_kernel(const float* __restrict__ ema_mean,
                                const float* __restrict__ ema_sq,
                                const float* __restrict__ ema_out,
                                const float* __restrict__ p_ltau,
                                const float* __restrict__ p_lbu,
                                const float* __restrict__ p_lbd,
                                const float* __restrict__ p_lg,
                                const float* __restrict__ p_lwg,
                                const float* __restrict__ p_lwl,
                                float* __restrict__ ws,
                                int nchunks, float invN) {
    __shared__ float red[THREADS];
    __shared__ float sc[4];
    const int tid = threadIdx.x;

    // ||ema_out||^2
    float acc = 0.f;
    for (int d = tid; d < DCH; d += THREADS) { float e = ema_out[d]; acc += e * e; }
    red[tid] = acc; __syncthreads();
    for (int s = THREADS / 2; s > 0; s >>= 1) {
        if (tid < s) red[tid] += red[tid + s];
        __syncthreads();
    }
    if (tid == 0) {
        sc[3] = 1.0f / fmaxf(sqrtf(red[0]), 1e-12f);      // inv_norm
        float lwg = *p_lwg, lwl = *p_lwl;
        float m = fmaxf(lwg, lwl);
        float eg = expf(lwg - m), el = expf(lwl - m);
        float wg = eg / (eg + el);
        sc[0] = wg; sc[1] = 1.0f - wg;
        sc[2] = log1pf(expf(*p_lg));                      // gamma = softplus
        float* o = ws + (size_t)nchunks * 2 * DCH + 3 * DCH;
        o[0] = log1pf(expf(*p_lbu));                      // beta_up
        o[1] = log1pf(expf(*p_lbd));                      // beta_dn
        o[2] = expf(*p_ltau);                             // tau
    }
    __syncthreads();
    const float wg = sc[0], wl = sc[1], gamma = sc[2], inv_norm = sc[3];

    float* A  = ws + (size_t)nchunks * 2 * DCH;
    float* Bc = A + DCH;
    float* En = A + 2 * DCH;
    for (int d = tid; d < DCH; d += THREADS) {
        float s = 0.f, q = 0.f;
        for (int c = 0; c < nchunks; ++c) {               // fixed order: deterministic
            const float* p = ws + (size_t)c * 2 * DCH;
            s += p[d];
            q += p[DCH + d];
        }
        float mu    = s * invN;
        float var_l = fmaxf(q * invN - mu * mu, 1e-4f);
        float inv_l = 1.0f / (sqrtf(var_l) + 1e-5f);
        float em    = ema_mean[d];
        float var_g = fmaxf(ema_sq[d] - em * em, 1e-4f);
        float inv_g = 1.0f / (sqrtf(var_g) + 1e-5f);
        A[d]  = gamma * (wg * inv_g + wl * inv_l);        // gamma*z = x*A - B
        Bc[d] = gamma * (wg * em * inv_g + wl * mu * inv_l);
        En[d] = ema_out[d] * inv_norm;
    }
}

// ---------------------------------------------------------------------------
// Kernel 3: one block per (b,t) row. Wave 0 DMAs the 24KB coefficient table
// (a|b|ema_n) into LDS via the Tensor Data Mover, waits TENSORcnt, barrier.
// All threads hold their 8 gelu values in registers, block-reduce
// (dot(g, ema_n), ||g||^2), then apply gates and NT-store the row.
// ---------------------------------------------------------------------------
__global__ void gelu_gate_kernel(const float* __restrict__ x,
                                 float* __restrict__ out,
                                 const float* __restrict__ wsChan,
                                 unsigned long long wsChanAddr) {
    __shared__ float s_chan[3 * DCH];     // a | b | ema_n
    __shared__ float s_dot[THREADS];
    __shared__ float s_sq[THREADS];

    const int tid = threadIdx.x;
    const int row = blockIdx.x;

#ifdef HAVE_TDM
    if (tid < 32) {
        const u32 lds_addr = (u32)(uintptr_t)(&s_chan[0]);   // low 32 bits = LDS offset
        const unsigned long long ga = wsChanAddr;
        const u32 nfl = 3 * DCH;                             // 6144 elements, 4B each
        v4u g0 = { 1u,                                       // count=1, user descriptor
                   lds_addr,
                   (u32)ga,
                   ((u32)(ga >> 32) & 0x01FFFFFFu) | 0x80000000u }; // addr[56:32] | type=2
        v8i g1 = { (int)0x00020000u,                         // data_size=4B
                   (int)((nfl & 0xFFFFu) << 16),             // tensor_dim0[15:0]
                   (int)((nfl >> 16) | (1u << 16)),          // tensor_dim0[31:16] | tensor_dim1=1
                   (int)((nfl & 0xFFFFu) << 16),             // tile_dim0 = 6144
                   0,                                        // tile_dim1=0 (1-D), tile_dim2=0
                   (int)nfl,                                 // tensor_dim0_stride
                   0, 0 };
        v4i gz = { 0, 0, 0, 0 };
#if __clang_major__ >= 23
        v8i gz8 = { 0, 0, 0, 0, 0, 0, 0, 0 };
        __builtin_amdgcn_tensor_load_to_lds(g0, g1, gz, gz, gz8, 0);
#else
        __builtin_amdgcn_tensor_load_to_lds(g0, g1, gz, gz, 0);
#endif
        __builtin_amdgcn_s_wait_tensorcnt(0);
    }
#else
    for (int i = tid; i < 3 * DCH; i += THREADS) s_chan[i] = wsChan[i];
#endif

    // Load row (overlaps the DMA), compute gelu in registers.
    const v4f* xr = (const v4f*)(x + (size_t)row * DCH);
    const v4f xa = xr[tid];
    const v4f xb = xr[tid + THREADS];
    float xs[8] = { xa.x, xa.y, xa.z, xa.w, xb.x, xb.y, xb.z, xb.w };
    float g[8];
    float sq = 0.f;
#pragma unroll
    for (int j = 0; j < 8; ++j) { g[j] = gelu_f(xs[j]); sq += g[j] * g[j]; }

    __syncthreads();   // LDS table valid for all waves

    const v4f* En4 = (const v4f*)(s_chan + 2 * DCH);
    const v4f e0 = En4[tid], e1 = En4[tid + THREADS];
    float dot = g[0]*e0.x + g[1]*e0.y + g[2]*e0.z + g[3]*e0.w
              + g[4]*e1.x + g[5]*e1.y + g[6]*e1.z + g[7]*e1.w;

    s_dot[tid] = dot; s_sq[tid] = sq;
    __syncthreads();
    for (int s = THREADS / 2; s > 0; s >>= 1) {
        if (tid < s) { s_dot[tid] += s_dot[tid + s]; s_sq[tid] += s_sq[tid + s]; }
        __syncthreads();
    }
    const float dotT = s_dot[0];
    const float sqT  = s_sq[0];

    const float beta_up = wsChan[3 * DCH + 0];
    const float beta_dn = wsChan[3 * DCH + 1];
    const float tau     = wsChan[3 * DCH + 2];

    const float nrm = fmaxf(sqrtf(sqT), 1e-12f);
    const float cs  = fminf(fmaxf(dotT / nrm, -1.0f), 1.0f);
    const float gate_cos = expf(-tau * cs);

    const v4f* A4 = (const v4f*)(s_chan);
    const v4f* B4 = (const v4f*)(s_chan + DCH);
    const v4f a0 = A4[tid], a1 = A4[tid + THREADS];
    const v4f b0 = B4[tid], b1 = B4[tid + THREADS];
    float av[8] = { a0.x, a0.y, a0.z, a0.w, a1.x, a1.y, a1.z, a1.w };
    float bv[8] = { b0.x, b0.y, b0.z, b0.w, b1.x, b1.y, b1.z, b1.w };

    float o[8];
#pragma unroll
    for (int j = 0; j < 8; ++j) {
        float u = tanhf(xs[j] * av[j] - bv[j]);           // tanh(gamma*z)
        float gate = 1.0f + beta_up * fmaxf(u, 0.0f) - beta_dn * fmaxf(-u, 0.0f);
        gate = fminf(fmaxf(gate, 0.05f), 8.0f);
        o[j] = g[j] * gate * gate_cos;
    }
    v4f* orow = (v4f*)(out + (size_t)row * DCH);
    v4f o0 = { o[0], o[1], o[2], o[3] };
    v4f o1 = { o[4], o[5], o[6], o[7] };
    __builtin_nontemporal_store(o0, &orow[tid]);
    __builtin_nontemporal_store(o1, &orow[tid + THREADS]);
}

// ---------------------------------------------------------------------------
extern "C" void kernel_launch(void* const* d_in, const int* in_sizes, int n_in,
                              void* d_out, int out_size, void* d_ws, size_t ws_size,
                              hipStream_t stream) {
    const float* x        = (const float*)d_in[0];
    const float* ema_mean = (const float*)d_in[1];
    const float* ema_sq   = (const float*)d_in[2];
    const float* ema_out  = (const float*)d_in[3];
    // d_in[4] = logit_decay (unused by reference output)
    const float* p_ltau   = (const float*)d_in[5];
    const float* p_lbu    = (const float*)d_in[6];
    const float* p_lbd    = (const float*)d_in[7];
    const float* p_lg     = (const float*)d_in[8];
    const float* p_lwg    = (const float*)d_in[9];
    const float* p_lwl    = (const float*)d_in[10];
    float* out = (float*)d_out;
    float* ws  = (float*)d_ws;

    const int rows = in_sizes[0] / DCH;      // B*T = 16384

    // Pick chunk count that fits ws and divides rows (deterministic per config).
    int nchunks = 128;
    while (nchunks > 1 &&
           (((size_t)nchunks * 2 * DCH + 3 * DCH + 8) * sizeof(float) > ws_size ||
            (rows % nchunks) != 0)) {
        nchunks >>= 1;
    }
    const int rpc = rows / nchunks;

    dim3 g1(DCH / (THREADS * 4), nchunks);
    colstats_kernel<<<g1, THREADS, 0, stream>>>(x, ws, rpc);

    finalize_kernel<<<1, THREADS, 0, stream>>>(ema_mean, ema_sq, ema_out,
                                               p_ltau, p_lbu, p_lbd, p_lg, p_lwg, p_lwl,
                                               ws, nchunks, 1.0f / (float)rows);

    const float* wsChan = ws + (size_t)nchunks * 2 * DCH;
    unsigned long long wsChanAddr = (unsigned long long)(uintptr_t)wsChan;
    gelu_gate_kernel<<<rows, THREADS, 0, stream>>>(x, out, wsChan, wsChanAddr);
}